// HeteroGAT_7215545058022
// MI455X (gfx1250) — compile-verified
//
#include <hip/hip_runtime.h>
#include <math.h>

// ---------------------------------------------------------------------------
// HeteroGAT forward for MI455X (gfx1250, wave32).
// Dense transforms: LDS-staged V_WMMA_F32_16X16X4_F32 (f32-exact path).
// Staging uses gfx1250 async global->LDS copies (guarded, with fallback).
// Edge phase: 3-pass segment softmax, logits cached, reciprocal-normalized.
// ---------------------------------------------------------------------------

#define HIDDEN 64
#define HEADS 4
#define HD 16
#define NEG 0.2f
#define EPSBN 1e-5f

typedef __attribute__((ext_vector_type(2))) float v2f;
typedef __attribute__((ext_vector_type(8))) float v8f;
typedef __attribute__((ext_vector_type(4))) int v4i;
typedef __attribute__((address_space(1))) v4i* gv4i_p;  // global
typedef __attribute__((address_space(3))) v4i* lv4i_p;  // LDS

__device__ __forceinline__ float lrelu(float x) { return x > 0.f ? x : NEG * x; }

__device__ __forceinline__ void atomicMaxF(float* addr, float v) {
  if (v >= 0.f) atomicMax((int*)addr, __float_as_int(v));
  else          atomicMin((unsigned int*)addr, __float_as_uint(v));
}

// 16-byte global -> LDS copy; async-to-LDS on gfx1250 toolchains that expose it
__device__ __forceinline__ void cp16(float* lds, const float* g) {
#if __has_builtin(__builtin_amdgcn_global_load_async_to_lds_b128)
  __builtin_amdgcn_global_load_async_to_lds_b128((gv4i_p)(g), (lv4i_p)(lds), 0, 0);
#else
  *(float4*)lds = *(const float4*)g;
#endif
}

__device__ __forceinline__ void cp16_flush() {
#if __has_builtin(__builtin_amdgcn_global_load_async_to_lds_b128)
#if __has_builtin(__builtin_amdgcn_s_wait_asynccnt)
  __builtin_amdgcn_s_wait_asynccnt(0);
#else
  asm volatile("s_wait_asynccnt 0" ::: "memory");
#endif
#endif
}

// ---------------- WMMA GEMM:  out[n,64] = A[n,K] @ W[K,64] (+bias) ----------
// block = 128 threads = 4 waves; wave w owns the 16x16 tile at cols [16w,16w+16)
template <int K>
__global__ void k_gemm(const float* __restrict__ A, const float* __restrict__ Wm,
                       const float* __restrict__ bias, float* __restrict__ out, int n) {
  constexpr int LAS = K + 4;  // padded row stride: keeps 16B align, spreads banks
  __shared__ float lA[16 * LAS];
  __shared__ float lB[K * HIDDEN];
  const int tid = threadIdx.x;
  const int lane = tid & 31;
  const int wave = tid >> 5;
  const int row0 = blockIdx.x * 16;
  if (row0 >= n) return;
  const bool full = (row0 + 16 <= n);

  // stage A tile (16 x K), coalesced b128
  for (int f = tid; f < 16 * (K / 4); f += 128) {
    int r = f / (K / 4), cg = (f % (K / 4)) * 4;
    int gr = full ? (row0 + r) : min(row0 + r, n - 1);
    cp16(&lA[r * LAS + cg], &A[(size_t)gr * K + cg]);
  }
  // stage B panel (K x 64), coalesced b128
  for (int f = tid; f < K * 16; f += 128) {
    int r = f >> 4, cg = (f & 15) * 4;
    cp16(&lB[r * HIDDEN + cg], &Wm[(size_t)r * HIDDEN + cg]);
  }
  cp16_flush();
  __syncthreads();

  const int col0 = wave * 16;
  const int m = lane & 15;
  const int kk = (lane >> 4) << 1;  // {0,2}
  v8f c = {};
#pragma unroll
  for (int k0 = 0; k0 < K; k0 += 4) {
    v2f a = *(const v2f*)&lA[m * LAS + k0 + kk];  // ds_load_b64, conflict-free
    v2f b;
    b.x = lB[(k0 + kk) * HIDDEN + col0 + m];
    b.y = lB[(k0 + kk + 1) * HIDDEN + col0 + m];
    c = __builtin_amdgcn_wmma_f32_16x16x4_f32(false, a, false, b, (short)0, c, false, false);
  }
  const float bb = bias ? bias[col0 + m] : 0.f;
  const int half = (lane >> 4) * 8;
  if (full) {
#pragma unroll
    for (int r = 0; r < 8; ++r)
      out[(size_t)(row0 + half + r) * HIDDEN + col0 + m] = c[r] + bb;
  } else {
#pragma unroll
    for (int r = 0; r < 8; ++r) {
      int row = row0 + half + r;
      if (row < n) out[(size_t)row * HIDDEN + col0 + m] = c[r] + bb;
    }
  }
}

// per-head 16x16 relation transform: out[n,h,16] = A[n,h,:] @ rel[h]  (wave = head)
__global__ void k_headmm(const float* __restrict__ A, const float* __restrict__ rel,
                         float* __restrict__ out, int n) {
  constexpr int LAS = HIDDEN + 4;
  __shared__ float lA[16 * LAS];
  __shared__ float lR[HEADS * HD * HD];
  const int tid = threadIdx.x;
  const int lane = tid & 31;
  const int h = tid >> 5;
  const int row0 = blockIdx.x * 16;
  if (row0 >= n) return;
  const bool full = (row0 + 16 <= n);

  for (int f = tid; f < 16 * (HIDDEN / 4); f += 128) {
    int r = f >> 4, cg = (f & 15) * 4;
    int gr = full ? (row0 + r) : min(row0 + r, n - 1);
    cp16(&lA[r * LAS + cg], &A[(size_t)gr * HIDDEN + cg]);
  }
  for (int f = tid; f < HEADS * HD * HD / 4; f += 128)
    cp16(&lR[f * 4], &rel[f * 4]);
  cp16_flush();
  __syncthreads();

  const int m = lane & 15;
  const int kk = (lane >> 4) << 1;
  const float* Wh = &lR[h * HD * HD];
  v8f c = {};
#pragma unroll
  for (int k0 = 0; k0 < HD; k0 += 4) {
    v2f a = *(const v2f*)&lA[m * LAS + h * HD + k0 + kk];
    v2f b;
    b.x = Wh[(k0 + kk) * HD + m];
    b.y = Wh[(k0 + kk + 1) * HD + m];
    c = __builtin_amdgcn_wmma_f32_16x16x4_f32(false, a, false, b, (short)0, c, false, false);
  }
  const int half = (lane >> 4) * 8;
  if (full) {
#pragma unroll
    for (int r = 0; r < 8; ++r)
      out[(size_t)(row0 + half + r) * HIDDEN + h * HD + m] = c[r];
  } else {
#pragma unroll
    for (int r = 0; r < 8; ++r) {
      int row = row0 + half + r;
      if (row < n) out[(size_t)row * HIDDEN + h * HD + m] = c[r];
    }
  }
}

// ---------------- GAT attention scalars -------------------------------------
__global__ void k_att(const float* __restrict__ ht, const float* __restrict__ att_s,
                      const float* __restrict__ att_d, float* __restrict__ as_,
                      float* __restrict__ ad_, int n) {
  int i = blockIdx.x * blockDim.x + threadIdx.x;
  if (i >= n) return;
  const float4* row = (const float4*)(ht + (size_t)i * HIDDEN);
  const float4* s4 = (const float4*)att_s;
  const float4* d4 = (const float4*)att_d;
#pragma unroll
  for (int h = 0; h < HEADS; ++h) {
    float ss = 0.f, dd = 0.f;
#pragma unroll
    for (int q = 0; q < 4; ++q) {
      float4 r = row[h * 4 + q], a = s4[h * 4 + q], b = d4[h * 4 + q];
      ss += r.x * a.x + r.y * a.y + r.z * a.z + r.w * a.w;
      dd += r.x * b.x + r.y * b.y + r.z * b.z + r.w * b.w;
    }
    as_[(size_t)i * 4 + h] = ss;
    ad_[(size_t)i * 4 + h] = dd;
  }
}

__global__ void k_self_max(const float* __restrict__ as_, const float* __restrict__ ad_,
                           float* __restrict__ m, int n) {
  int i = blockIdx.x * blockDim.x + threadIdx.x;
  if (i >= n) return;
#pragma unroll
  for (int h = 0; h < 4; ++h)
    m[(size_t)i * 4 + h] = lrelu(as_[(size_t)i * 4 + h] + ad_[(size_t)i * 4 + h]);
}

// store GAT logits + running segment max
__global__ void k_gat_logit_max(const int* __restrict__ src, const int* __restrict__ dst,
                                const float* __restrict__ as_, const float* __restrict__ ad_,
                                float* __restrict__ wlog, float* __restrict__ m, int E) {
  int e = blockIdx.x * blockDim.x + threadIdx.x;
  if (e >= E) return;
  int s = src[e], d = dst[e];
  float4 a = *(const float4*)(as_ + (size_t)s * 4);
  float4 b = *(const float4*)(ad_ + (size_t)d * 4);
  float4 l;
  l.x = lrelu(a.x + b.x); l.y = lrelu(a.y + b.y);
  l.z = lrelu(a.z + b.z); l.w = lrelu(a.w + b.w);
  *(float4*)(wlog + (size_t)e * 4) = l;
  atomicMaxF(&m[(size_t)d * 4 + 0], l.x);
  atomicMaxF(&m[(size_t)d * 4 + 1], l.y);
  atomicMaxF(&m[(size_t)d * 4 + 2], l.z);
  atomicMaxF(&m[(size_t)d * 4 + 3], l.w);
}

__global__ void k_self_sum(const float* __restrict__ as_, const float* __restrict__ ad_,
                           const float* __restrict__ m, float* __restrict__ s, int n) {
  int i = blockIdx.x * blockDim.x + threadIdx.x;
  if (i >= n) return;
#pragma unroll
  for (int h = 0; h < 4; ++h) {
    float v = lrelu(as_[(size_t)i * 4 + h] + ad_[(size_t)i * 4 + h]);
    s[(size_t)i * 4 + h] = __expf(v - m[(size_t)i * 4 + h]);
  }
}

// shared (GAT + HGT): w = exp(logit - m[d]); ssum[d] += w   (streams wlog)
__global__ void k_expsum(const int* __restrict__ dst, float* __restrict__ wlog,
                         const float* __restrict__ m, float* __restrict__ ssum, int E) {
  int e = blockIdx.x * blockDim.x + threadIdx.x;
  if (e >= E) return;
  int d = dst[e];
  float4 l = *(const float4*)(wlog + (size_t)e * 4);
  float4 mm = *(const float4*)(m + (size_t)d * 4);
  float4 w;
  w.x = __expf(l.x - mm.x); w.y = __expf(l.y - mm.y);
  w.z = __expf(l.z - mm.z); w.w = __expf(l.w - mm.w);
  *(float4*)(wlog + (size_t)e * 4) = w;
  atomicAdd(&ssum[(size_t)d * 4 + 0], w.x);
  atomicAdd(&ssum[(size_t)d * 4 + 1], w.y);
  atomicAdd(&ssum[(size_t)d * 4 + 2], w.z);
  atomicAdd(&ssum[(size_t)d * 4 + 3], w.w);
}

__global__ void k_recip(float* __restrict__ s, size_t cnt) {
  size_t i = (size_t)blockIdx.x * blockDim.x + threadIdx.x;
  if (i < cnt) s[i] = 1.f / (s[i] + 1e-16f);
}

// wave-per-edge weighted scatter: acc[d,:] += feat[s,:] * w[e,h] * sinv[d,h]
__global__ void k_edge_scatter(const int* __restrict__ src, const int* __restrict__ dst,
                               const float* __restrict__ feat, const float* __restrict__ w,
                               const float* __restrict__ sinv, float* __restrict__ acc, int E) {
  int gid = blockIdx.x * blockDim.x + threadIdx.x;
  int e = gid >> 5;
  int lane = threadIdx.x & 31;
  if (e >= E) return;
  int s = src[e], d = dst[e];
  __builtin_prefetch(feat + (size_t)s * HIDDEN, 0, 0);
#pragma unroll
  for (int i = 0; i < 2; ++i) {
    int c = lane + 32 * i;
    int h = c >> 4;
    float alpha = w[(size_t)e * 4 + h] * sinv[(size_t)d * 4 + h];
    atomicAdd(&acc[(size_t)d * HIDDEN + c], feat[(size_t)s * HIDDEN + c] * alpha);
  }
}

__global__ void k_self_scatter(const float* __restrict__ as_, const float* __restrict__ ad_,
                               const float* __restrict__ m, const float* __restrict__ sinv,
                               const float* __restrict__ feat, float* __restrict__ acc, int n) {
  int gid = blockIdx.x * blockDim.x + threadIdx.x;
  int i = gid >> 5;
  int lane = threadIdx.x & 31;
  if (i >= n) return;
#pragma unroll
  for (int j = 0; j < 2; ++j) {
    int c = lane + 32 * j;
    int h = c >> 4;
    float v = lrelu(as_[(size_t)i * 4 + h] + ad_[(size_t)i * 4 + h]);
    float alpha = __expf(v - m[(size_t)i * 4 + h]) * sinv[(size_t)i * 4 + h];
    acc[(size_t)i * HIDDEN + c] += feat[(size_t)i * HIDDEN + c] * alpha;
  }
}

__global__ void k_init_bias3(float* __restrict__ acc, const float* __restrict__ b0,
                             const float* __restrict__ b1, const float* __restrict__ b2, int n) {
  size_t idx = (size_t)blockIdx.x * blockDim.x + threadIdx.x;
  if (idx >= (size_t)n * HIDDEN) return;
  int c = idx & 63;
  acc[idx] = b0[c] + b1[c] + b2[c];
}

__global__ void k_fill(float* __restrict__ p, float v, size_t cnt) {
  size_t i = (size_t)blockIdx.x * blockDim.x + threadIdx.x;
  if (i < cnt) p[i] = v;
}

// ---------------- BatchNorm -------------------------------------------------
__global__ void k_bn_stats(const float* __restrict__ x, float* __restrict__ sums, int n) {
  int c = threadIdx.x;  // 64 threads, one column each
  float s = 0.f, s2 = 0.f;
  for (int r = blockIdx.x; r < n; r += gridDim.x) {
    float v = x[(size_t)r * HIDDEN + c];
    s += v; s2 += v * v;
  }
  atomicAdd(&sums[c], s);
  atomicAdd(&sums[64 + c], s2);
}

__global__ void k_bn_finalize(const float* __restrict__ sums, const float* __restrict__ g,
                              const float* __restrict__ b, float* __restrict__ bnp, int n) {
  int c = threadIdx.x;
  if (c >= 64) return;
  float mu = sums[c] / n;
  float var = sums[64 + c] / n - mu * mu;
  float sc = g[c] * rsqrtf(var + EPSBN);
  bnp[c] = sc;
  bnp[64 + c] = b[c] - mu * sc;
}

__global__ void k_apply_bn_lrelu(const float* __restrict__ acc, const float* __restrict__ bnp,
                                 float* __restrict__ out, int n) {
  size_t idx = (size_t)blockIdx.x * blockDim.x + threadIdx.x;
  if (idx >= (size_t)n * HIDDEN) return;
  int c = idx & 63;
  out[idx] = lrelu(acc[idx] * bnp[c] + bnp[64 + c]);
}

__global__ void k_apply_layer(float* __restrict__ h, const float* __restrict__ acc,
                              const float* __restrict__ proj, const float* __restrict__ bnp, int n) {
  size_t idx = (size_t)blockIdx.x * blockDim.x + threadIdx.x;
  if (idx >= (size_t)n * HIDDEN) return;
  int c = idx & 63;
  h[idx] = lrelu(h[idx] + acc[idx] * bnp[c] + bnp[64 + c] + proj[idx]);
}

__global__ void k_apply_hgt(float* __restrict__ h, const float* __restrict__ xn,
                            const float* __restrict__ bnp, int n) {
  size_t idx = (size_t)blockIdx.x * blockDim.x + threadIdx.x;
  if (idx >= (size_t)n * HIDDEN) return;
  int c = idx & 63;
  h[idx] = lrelu(h[idx] + xn[idx] * bnp[c] + bnp[64 + c]);
}

__global__ void k_gelu(float* __restrict__ x, int n) {
  size_t idx = (size_t)blockIdx.x * blockDim.x + threadIdx.x;
  if (idx >= (size_t)n * HIDDEN) return;
  float v = x[idx];
  float u = 0.7978845608028654f * (v + 0.044715f * v * v * v);
  x[idx] = 0.5f * v * (1.f + tanhf(u));
}

__global__ void k_combine_skip(const float* __restrict__ o, const float* __restrict__ h,
                               const float* __restrict__ skip, float* __restrict__ dst, int n) {
  size_t idx = (size_t)blockIdx.x * blockDim.x + threadIdx.x;
  if (idx >= (size_t)n * HIDDEN) return;
  float g = 1.f / (1.f + __expf(-skip[0]));
  dst[idx] = g * o[idx] + (1.f - g) * h[idx];
}

// ---------------- HGT edge kernels ------------------------------------------
__global__ void k_hgt_logit_max(const int* __restrict__ src, const int* __restrict__ dst,
                                const float* __restrict__ q, const float* __restrict__ kr,
                                const float* __restrict__ prel, float* __restrict__ wlog,
                                float* __restrict__ m, int E) {
  int e = blockIdx.x * blockDim.x + threadIdx.x;
  if (e >= E) return;
  int s = src[e], d = dst[e];
  const float4* qr = (const float4*)(q + (size_t)d * HIDDEN);
  const float4* kv = (const float4*)(kr + (size_t)s * HIDDEN);
#pragma unroll
  for (int h = 0; h < 4; ++h) {
    float dot = 0.f;
#pragma unroll
    for (int j = 0; j < 4; ++j) {
      float4 a = qr[h * 4 + j], b = kv[h * 4 + j];
      dot += a.x * b.x + a.y * b.y + a.z * b.z + a.w * b.w;
    }
    float l = dot * prel[h] * 0.25f;  // 1/sqrt(HEAD_DIM)
    wlog[(size_t)e * 4 + h] = l;
    atomicMaxF(&m[(size_t)d * 4 + h], l);
  }
}

__global__ void k_final(const float* __restrict__ h, const float* __restrict__ Wl,
                        const float* __restrict__ bl, float* __restrict__ y, int n) {
  int i = blockIdx.x * blockDim.x + threadIdx.x;
  if (i >= n) return;
  const float4* r = (const float4*)(h + (size_t)i * HIDDEN);
  const float4* w = (const float4*)Wl;
  float acc = 0.f;
#pragma unroll
  for (int j = 0; j < 16; ++j) {
    float4 a = r[j], b = w[j];
    acc += a.x * b.x + a.y * b.y + a.z * b.z + a.w * b.w;
  }
  y[i] = acc + bl[0];
}

// ---------------------------------------------------------------------------
static inline int gdiv(long a, long b) { return (int)((a + b - 1) / b); }

extern "C" void kernel_launch(void* const* d_in, const int* in_sizes, int n_in,
                              void* d_out, int out_size, void* d_ws, size_t ws_size,
                              hipStream_t stream) {
  int icur = 0;
  const float* x = (const float*)d_in[icur++];
  const int* ei_line = (const int*)d_in[icur++];
  const int* ei_region = (const int*)d_in[icur++];
  const int* ei_diag = (const int*)d_in[icur++];
  auto F = [&]() { return (const float*)d_in[icur++]; };

  // params flattened in jax tree_leaves order (dict keys sorted; 'W' < 'a' ASCII)
  const float *bn_b[3], *bn_g[3];
  for (int l = 0; l < 3; ++l) { bn_b[l] = F(); bn_g[l] = F(); }
  const float *bnh_b = F(), *bnh_g = F();
  struct GP { const float *W, *att_dst, *att_src, *b; };
  GP gat[3][3];  // [layer][type: 0=diag,1=line,2=region]
  for (int l = 0; l < 3; ++l)
    for (int t = 0; t < 3; ++t) {
      gat[l][t].W = F(); gat[l][t].att_dst = F(); gat[l][t].att_src = F(); gat[l][t].b = F();
    }
  const float *Wa = F(), *Wk = F(), *Wq = F(), *Wv = F();
  const float* a_rel[3]; for (int t = 0; t < 3; ++t) a_rel[t] = F();
  const float *ba = F(), *bk = F(), *bq = F(), *bv = F();
  const float* m_rel[3]; for (int t = 0; t < 3; ++t) m_rel[t] = F();
  const float* p_rel[3]; for (int t = 0; t < 3; ++t) p_rel[t] = F();
  const float* skip = F();
  const float *linW = F(), *linb = F();
  const float *projW[2], *projb[2];
  for (int l = 0; l < 2; ++l) { projW[l] = F(); projb[l] = F(); }

  const int n = in_sizes[0] / 16;
  const int* srcs[3] = { ei_diag, ei_line, ei_region };
  int E[3] = { in_sizes[3] / 2, in_sizes[1] / 2, in_sizes[2] / 2 };
  long Etot = (long)E[0] + E[1] + E[2];

  // workspace carve
  float* P = (float*)d_ws;
  auto take = [&](size_t cnt) { float* r = P; P += cnt; return r; };
  size_t NH = (size_t)n * HIDDEN;
  float* buf_h = take(NH);
  float* buf_acc = take(NH);
  float* hts[3] = { take(NH), take(NH), take(NH) };
  float* buf_x0 = take(NH);
  float* buf_x1 = take(NH);
  float* asrc[3]; float* adst[3];
  for (int t = 0; t < 3; ++t) { asrc[t] = take((size_t)n * 4); adst[t] = take((size_t)n * 4); }
  float* mbuf = take((size_t)n * 4);
  float* sbuf = take((size_t)n * 4);
  float* wbuf = take((size_t)Etot * 4);
  float* sums = take(128);
  float* bnp = take(128);
  (void)ws_size; (void)n_in; (void)out_size;

  const int gN = gdiv(n, 256);
  const int gNH = gdiv((long)NH, 256);
  const int gT = gdiv(n, 16);
  const int gNw = gdiv((long)n * 32, 256);
  const int gH4 = gdiv((long)n * 4, 256);

  // ---- GAT layers ----
  const float* layer_in = x;
  for (int l = 0; l < 3; ++l) {
    for (int t = 0; t < 3; ++t) {
      if (l == 0) k_gemm<16><<<gT, 128, 0, stream>>>(layer_in, gat[l][t].W, nullptr, hts[t], n);
      else        k_gemm<64><<<gT, 128, 0, stream>>>(layer_in, gat[l][t].W, nullptr, hts[t], n);
      k_att<<<gN, 256, 0, stream>>>(hts[t], gat[l][t].att_src, gat[l][t].att_dst, asrc[t], adst[t], n);
    }
    k_init_bias3<<<gNH, 256, 0, stream>>>(buf_acc, gat[l][0].b, gat[l][1].b, gat[l][2].b, n);
    for (int t = 0; t < 3; ++t) {
      const int* s_ = srcs[t];
      const int* d_ = srcs[t] + E[t];
      const int gE = gdiv(E[t], 256);
      const int gEw = gdiv((long)E[t] * 32, 256);
      k_self_max<<<gN, 256, 0, stream>>>(asrc[t], adst[t], mbuf, n);
      k_gat_logit_max<<<gE, 256, 0, stream>>>(s_, d_, asrc[t], adst[t], wbuf, mbuf, E[t]);
      k_self_sum<<<gN, 256, 0, stream>>>(asrc[t], adst[t], mbuf, sbuf, n);
      k_expsum<<<gE, 256, 0, stream>>>(d_, wbuf, mbuf, sbuf, E[t]);
      k_recip<<<gH4, 256, 0, stream>>>(sbuf, (size_t)n * 4);
      k_edge_scatter<<<gEw, 256, 0, stream>>>(s_, d_, hts[t], wbuf, sbuf, buf_acc, E[t]);
      k_self_scatter<<<gNw, 256, 0, stream>>>(asrc[t], adst[t], mbuf, sbuf, hts[t], buf_acc, n);
    }
    k_fill<<<1, 128, 0, stream>>>(sums, 0.f, 128);
    k_bn_stats<<<256, 64, 0, stream>>>(buf_acc, sums, n);
    k_bn_finalize<<<1, 64, 0, stream>>>(sums, bn_g[l], bn_b[l], bnp, n);
    if (l == 0) {
      k_apply_bn_lrelu<<<gNH, 256, 0, stream>>>(buf_acc, bnp, buf_h, n);
    } else {
      k_gemm<16><<<gT, 128, 0, stream>>>(x, projW[l - 1], projb[l - 1], buf_x0, n);
      k_apply_layer<<<gNH, 256, 0, stream>>>(buf_h, buf_acc, buf_x0, bnp, n);
    }
    layer_in = buf_h;
  }

  // ---- HGT ----
  k_gemm<64><<<gT, 128, 0, stream>>>(buf_h, Wk, bk, hts[0], n);  // k
  k_gemm<64><<<gT, 128, 0, stream>>>(buf_h, Wq, bq, hts[1], n);  // q
  k_gemm<64><<<gT, 128, 0, stream>>>(buf_h, Wv, bv, hts[2], n);  // v
  k_fill<<<gH4, 256, 0, stream>>>(mbuf, -INFINITY, (size_t)n * 4);
  k_fill<<<gH4, 256, 0, stream>>>(sbuf, 0.f, (size_t)n * 4);
  k_fill<<<gNH, 256, 0, stream>>>(buf_acc, 0.f, NH);
  long base = 0;
  for (int t = 0; t < 3; ++t) {
    k_headmm<<<gT, 128, 0, stream>>>(hts[0], a_rel[t], buf_x0, n);  // kr
    k_hgt_logit_max<<<gdiv(E[t], 256), 256, 0, stream>>>(srcs[t], srcs[t] + E[t], hts[1], buf_x0,
                                                         p_rel[t], wbuf + base * 4, mbuf, E[t]);
    base += E[t];
  }
  base = 0;
  for (int t = 0; t < 3; ++t) {
    k_expsum<<<gdiv(E[t], 256), 256, 0, stream>>>(srcs[t] + E[t], wbuf + base * 4, mbuf, sbuf, E[t]);
    base += E[t];
  }
  k_recip<<<gH4, 256, 0, stream>>>(sbuf, (size_t)n * 4);
  base = 0;
  for (int t = 0; t < 3; ++t) {
    k_headmm<<<gT, 128, 0, stream>>>(hts[2], m_rel[t], buf_x0, n);  // vr
    k_edge_scatter<<<gdiv((long)E[t] * 32, 256), 256, 0, stream>>>(srcs[t], srcs[t] + E[t], buf_x0,
                                                                   wbuf + base * 4, sbuf, buf_acc, E[t]);
    base += E[t];
  }
  k_gelu<<<gNH, 256, 0, stream>>>(buf_acc, n);
  k_gemm<64><<<gT, 128, 0, stream>>>(buf_acc, Wa, ba, buf_x0, n);
  k_combine_skip<<<gNH, 256, 0, stream>>>(buf_x0, buf_h, skip, buf_x1, n);
  k_fill<<<1, 128, 0, stream>>>(sums, 0.f, 128);
  k_bn_stats<<<256, 64, 0, stream>>>(buf_x1, sums, n);
  k_bn_finalize<<<1, 64, 0, stream>>>(sums, bnh_g, bnh_b, bnp, n);
  k_apply_hgt<<<gNH, 256, 0, stream>>>(buf_h, buf_x1, bnp, n);
  k_final<<<gN, 256, 0, stream>>>(buf_h, linW, linb, (float*)d_out, n);
}